// PLIFNode1D_26499948216579
// MI455X (gfx1250) — compile-verified
//
#include <hip/hip_runtime.h>
#include <cstddef>

typedef float v4f __attribute__((ext_vector_type(4)));
typedef int   v4i __attribute__((ext_vector_type(4)));

// Address-space-qualified vector pointees matching the builtin's signature:
//   __builtin_amdgcn_global_load_async_to_lds_b128(int4 AS1*, int4 AS3*, Ii, Ii)
typedef __attribute__((address_space(1))) v4i gbl_v4i;
typedef __attribute__((address_space(3))) v4i lds_v4i;

namespace {
constexpr int   kB     = 128;
constexpr int   kT     = 256;
constexpr int   kC     = 2048;
constexpr int   kC4    = kC / 4;      // 512 float4 columns per (b,t) row
constexpr int   kBlock = 256;         // 8 wave32 waves per block
constexpr int   kTS    = 4;           // timesteps staged per tile
constexpr int   kTiles = kT / kTS;    // 64 tiles over time
constexpr float kVth   = 0.5f;
}

// Device pass + toolchain supports gfx1250 async global->LDS loads?
#if defined(__HIP_DEVICE_COMPILE__) && \
    __has_builtin(__builtin_amdgcn_global_load_async_to_lds_b128)
#define PLIF_USE_ASYNC 1
#else
#define PLIF_USE_ASYNC 0
#endif

__device__ __forceinline__ v4f plif_step(v4f& v, const v4f xt, const v4f decay) {
  // v = v*decay + x_t  (contracts to v_fma)
  v = v * decay + xt;
  v4f s;
  s.x = (v.x > kVth) ? 1.0f : 0.0f;
  s.y = (v.y > kVth) ? 1.0f : 0.0f;
  s.z = (v.z > kVth) ? 1.0f : 0.0f;
  s.w = (v.w > kVth) ? 1.0f : 0.0f;
  v = v - s * kVth;  // soft reset by threshold on spike
  return s;
}

__global__ void __launch_bounds__(kBlock)
plif_scan_kernel(const float* __restrict__ x,
                 const float* __restrict__ decay_param,
                 float* __restrict__ out) {
#if PLIF_USE_ASYNC
  // Per-lane private staging columns; double buffered. 2*4*256*16B = 32 KB.
  __shared__ v4f stage[2][kTS][kBlock];
#endif
  const int tid = threadIdx.x;
  const int gid = blockIdx.x * kBlock + tid;
  const int b   = gid / kC4;
  const int c4  = gid % kC4;

  // decay = sigmoid(decay_param), constant over the scan
  const v4f dp = reinterpret_cast<const v4f*>(decay_param)[c4];
  v4f decay;
  decay.x = 1.0f / (1.0f + __expf(-dp.x));
  decay.y = 1.0f / (1.0f + __expf(-dp.y));
  decay.z = 1.0f / (1.0f + __expf(-dp.z));
  decay.w = 1.0f / (1.0f + __expf(-dp.w));

  const v4f* __restrict__ xp =
      reinterpret_cast<const v4f*>(x) + (size_t)b * kT * kC4 + c4;
  v4f* __restrict__ op =
      reinterpret_cast<v4f*>(out) + (size_t)b * kT * kC4 + c4;

  v4f v = {0.0f, 0.0f, 0.0f, 0.0f};

#if PLIF_USE_ASYNC
  // Prime buffer 0: each lane DMAs its own kTS timesteps into its LDS column.
#pragma unroll
  for (int i = 0; i < kTS; ++i) {
    __builtin_amdgcn_global_load_async_to_lds_b128(
        (gbl_v4i*)(xp + (size_t)i * kC4),
        (lds_v4i*)&stage[0][i][tid],
        0, 0);
  }
  for (int k = 0; k < kTiles; ++k) {
    const int buf = k & 1;
    if (k + 1 < kTiles) {
      // Issue next tile into the other buffer, then wait until only those
      // kTS loads remain outstanding (async loads complete in order, so
      // tile k's data has landed in LDS).
#pragma unroll
      for (int i = 0; i < kTS; ++i) {
        __builtin_amdgcn_global_load_async_to_lds_b128(
            (gbl_v4i*)(xp + (size_t)((k + 1) * kTS + i) * kC4),
            (lds_v4i*)&stage[buf ^ 1][i][tid],
            0, 0);
      }
#if __has_builtin(__builtin_amdgcn_s_wait_asynccnt)
      __builtin_amdgcn_s_wait_asynccnt(kTS);
#else
      asm volatile("s_wait_asynccnt 4" ::: "memory");
#endif
    } else {
#if __has_builtin(__builtin_amdgcn_s_wait_asynccnt)
      __builtin_amdgcn_s_wait_asynccnt(0);
#else
      asm volatile("s_wait_asynccnt 0" ::: "memory");
#endif
    }
    // Each lane reads back only the LDS it DMA'd itself -> no barrier needed.
#pragma unroll
    for (int i = 0; i < kTS; ++i) {
      const v4f xt = stage[buf][i][tid];
      const v4f s  = plif_step(v, xt, decay);
      __builtin_nontemporal_store(s, op + (size_t)(k * kTS + i) * kC4);
    }
  }
#else
  // Fallback: direct streaming 128-bit loads with software prefetch
  // (global_prefetch_b8) and non-temporal accesses.
  for (int t = 0; t < kT; ++t) {
    if (t + 8 < kT) __builtin_prefetch(xp + (size_t)(t + 8) * kC4, 0, 0);
    const v4f xt = __builtin_nontemporal_load(xp + (size_t)t * kC4);
    const v4f s  = plif_step(v, xt, decay);
    __builtin_nontemporal_store(s, op + (size_t)t * kC4);
  }
#endif
}

extern "C" void kernel_launch(void* const* d_in, const int* in_sizes, int n_in,
                              void* d_out, int out_size, void* d_ws, size_t ws_size,
                              hipStream_t stream) {
  (void)in_sizes; (void)n_in; (void)d_ws; (void)ws_size; (void)out_size;
  const float* x           = (const float*)d_in[0];
  const float* decay_param = (const float*)d_in[1];
  float*       out         = (float*)d_out;

  const int total_threads = kB * kC4;                // 65536 lanes of 4 channels
  const int grid          = total_threads / kBlock;  // 256 blocks
  plif_scan_kernel<<<grid, kBlock, 0, stream>>>(x, decay_param, out);
}